// UnifiedRadialMLP_4569845203223
// MI455X (gfx1250) — compile-verified
//
#include <hip/hip_runtime.h>
#include <hip/hip_bf16.h>

#define E_TOT 150000
#define H 128
#define NLAY 8
#define TE 48          // edge rows per block (150000 % 48 == 0)
#define LDS_STRIDE 132 // padded row stride in floats -> conflict-free frag loads
#define EPS 1e-5f

typedef __attribute__((ext_vector_type(2))) float v2f;
typedef __attribute__((ext_vector_type(8))) float v8f;

static_assert(E_TOT % TE == 0, "tile must divide E");

__device__ __forceinline__ float silu_f(float x) {
    return x / (1.0f + __expf(-x));
}

// C(48x128) = A(48x128, LDS padded) * W(128x128 global row-major [n][k])^T + bias
// Wave w owns N columns [16w, 16w+16); loops 3 M-tiles of 16 rows.
// B fragments for the whole K range are preloaded into 32 v2f registers and
// reused across the 3 M-tiles.
template <bool TO_GLOBAL>
__device__ __forceinline__ void gemm48(const float* __restrict__ aBuf,  // LDS
                                       const float* __restrict__ W,    // global HxH
                                       const float* __restrict__ bias, // global H
                                       float* __restrict__ dst,        // LDS or global
                                       int lane, int wave) {
    const int nBase = wave * 16;
    const int nCol  = nBase + (lane & 15);
    const int kOff  = (lane >> 4) * 2; // lanes 0-15: K{0,1}; lanes 16-31: K{2,3}

    v2f bfrag[32];
#pragma unroll
    for (int j = 0; j < 32; ++j) {
        bfrag[j] = *(const v2f*)(W + nCol * H + kOff + 4 * j);
    }
    const float bv = bias[nCol];

#pragma unroll
    for (int mt = 0; mt < 3; ++mt) {
        const int mBase = mt * 16;
        v8f acc = {bv, bv, bv, bv, bv, bv, bv, bv};
        const float* aRow = aBuf + (mBase + (lane & 15)) * LDS_STRIDE + kOff;
#pragma unroll
        for (int j = 0; j < 32; ++j) {
            v2f afrag = *(const v2f*)(aRow + 4 * j);
            acc = __builtin_amdgcn_wmma_f32_16x16x4_f32(
                false, afrag, false, bfrag[j], (short)0, acc, false, false);
        }
        // C/D layout: VGPR i -> row mBase + i (+8 for lanes 16-31), col = lane&15
        const int row0 = mBase + ((lane >> 4) << 3);
        const int col  = lane & 15;
#pragma unroll
        for (int i = 0; i < 8; ++i) {
            if (TO_GLOBAL)
                dst[(size_t)(row0 + i) * H + nBase + col] = acc[i];
            else
                dst[(row0 + i) * LDS_STRIDE + nBase + col] = acc[i];
        }
    }
}

// LayerNorm + SiLU over rows of 128; 4 lanes per row, wave-local shuffles.
__device__ __forceinline__ void ln_silu(const float* __restrict__ src,
                                        float* __restrict__ dst,
                                        const float* __restrict__ w,
                                        const float* __restrict__ b, int tid) {
    if (tid < TE * 4) {
        const int row  = tid >> 2;
        const int part = tid & 3;
        const float* p = src + row * LDS_STRIDE + part * 32;
        float sum = 0.0f, sq = 0.0f;
#pragma unroll
        for (int i = 0; i < 32; ++i) {
            float v = p[i];
            sum += v;
            sq += v * v;
        }
        sum += __shfl_xor(sum, 1, 32);
        sq  += __shfl_xor(sq, 1, 32);
        sum += __shfl_xor(sum, 2, 32);
        sq  += __shfl_xor(sq, 2, 32);
        const float mean = sum * (1.0f / 128.0f);
        const float var  = sq * (1.0f / 128.0f) - mean * mean;
        const float inv  = rsqrtf(var + EPS);
        float* q = dst + row * LDS_STRIDE + part * 32;
#pragma unroll
        for (int i = 0; i < 32; ++i) {
            const int col = part * 32 + i;
            float v = (p[i] - mean) * inv * w[col] + b[col];
            q[i] = silu_f(v);
        }
    }
}

__global__ __launch_bounds__(256) void unified_radial_mlp_kernel(
    const float* __restrict__ x, const float* __restrict__ W1_cat,
    const float* __restrict__ b1_cat, const float* __restrict__ eln1w,
    const float* __restrict__ eln1b, const float* __restrict__ efc2w,
    const float* __restrict__ efc2b, const float* __restrict__ eln2w,
    const float* __restrict__ eln2b, const float* __restrict__ efc3w,
    const float* __restrict__ efc3b, const float* __restrict__ ln1w,
    const float* __restrict__ ln1b, const float* __restrict__ fc2w,
    const float* __restrict__ fc2b, const float* __restrict__ ln2w,
    const float* __restrict__ ln2b, const float* __restrict__ fc3w,
    const float* __restrict__ fc3b, float* __restrict__ out) {
    __shared__ float bufA[TE * LDS_STRIDE];
    __shared__ float bufB[TE * LDS_STRIDE];

    const int s    = blockIdx.y;      // 0 = edge stream, 1..8 = layer streams
    const int e0   = blockIdx.x * TE; // first edge row of tile
    const int tid  = threadIdx.x;
    const int lane = tid & 31;
    const int wave = tid >> 5;

    // per-stream parameter pointers
    const float *lw1, *lb1, *w2, *b2, *lw2, *lb2, *w3, *b3;
    if (s == 0) {
        lw1 = eln1w; lb1 = eln1b;
        w2 = efc2w;  b2 = efc2b;
        lw2 = eln2w; lb2 = eln2b;
        w3 = efc3w;  b3 = efc3b;
    } else {
        const int l = s - 1;
        lw1 = ln1w + l * H;     lb1 = ln1b + l * H;
        w2  = fc2w + (size_t)l * H * H; b2 = fc2b + l * H;
        lw2 = ln2w + l * H;     lb2 = ln2b + l * H;
        w3  = fc3w + (size_t)l * H * H; b3 = fc3b + l * H;
    }

    // load x tile (48 x 128) into bufA, coalesced float4
    for (int i = tid; i < TE * (H / 4); i += 256) {
        const int r  = i / (H / 4);
        const int c4 = i % (H / 4);
        float4 v = ((const float4*)(x + (size_t)(e0 + r) * H))[c4];
        *(float4*)&bufA[r * LDS_STRIDE + c4 * 4] = v;
    }
    __syncthreads();

    // stage 1: h = x @ W1[s].T + b1[s]
    gemm48<false>(bufA, W1_cat + (size_t)s * H * H, b1_cat + s * H, bufB, lane, wave);
    __syncthreads();
    ln_silu(bufB, bufA, lw1, lb1, tid);
    __syncthreads();

    // stage 2: h = silu(ln(h)) @ fc2.T + b2
    gemm48<false>(bufA, w2, b2, bufB, lane, wave);
    __syncthreads();
    ln_silu(bufB, bufA, lw2, lb2, tid);
    __syncthreads();

    // stage 3: out[s] = silu(ln(h)) @ fc3.T + b3  (direct to global)
    float* outBase = out + ((size_t)s * E_TOT + e0) * H;
    gemm48<true>(bufA, w3, b3, outBase, lane, wave);
}

extern "C" void kernel_launch(void* const* d_in, const int* in_sizes, int n_in,
                              void* d_out, int out_size, void* d_ws, size_t ws_size,
                              hipStream_t stream) {
    const float* x      = (const float*)d_in[0];
    const float* W1_cat = (const float*)d_in[1];
    const float* b1_cat = (const float*)d_in[2];
    const float* eln1w  = (const float*)d_in[3];
    const float* eln1b  = (const float*)d_in[4];
    const float* efc2w  = (const float*)d_in[5];
    const float* efc2b  = (const float*)d_in[6];
    const float* eln2w  = (const float*)d_in[7];
    const float* eln2b  = (const float*)d_in[8];
    const float* efc3w  = (const float*)d_in[9];
    const float* efc3b  = (const float*)d_in[10];
    const float* ln1w   = (const float*)d_in[11];
    const float* ln1b   = (const float*)d_in[12];
    const float* fc2w   = (const float*)d_in[13];
    const float* fc2b   = (const float*)d_in[14];
    const float* ln2w   = (const float*)d_in[15];
    const float* ln2b   = (const float*)d_in[16];
    const float* fc3w   = (const float*)d_in[17];
    const float* fc3b   = (const float*)d_in[18];
    float* out = (float*)d_out;

    dim3 grid(E_TOT / TE, 1 + NLAY);
    unified_radial_mlp_kernel<<<grid, 256, 0, stream>>>(
        x, W1_cat, b1_cat, eln1w, eln1b, efc2w, efc2b, eln2w, eln2b, efc3w,
        efc3b, ln1w, ln1b, fc2w, fc2b, ln2w, ln2b, fc3w, fc3b, out);
}